// WinToMeNeighborhoodAttention2D_39505109188968
// MI455X (gfx1250) — compile-verified
//
#include <hip/hip_runtime.h>

#define BATCH 2
#define HH 56
#define WW 56
#define CC 256
#define NHD 8
#define DD 32
#define KS 7
#define HW (HH * WW)        // 3136
#define TOK (BATCH * HW)    // 6272
#define N3C (3 * CC)        // 768
#define RPB ((2 * KS - 1) * (2 * KS - 1))  // 169

#define KC  32              // GEMM K-chunk staged per TDM transfer
#define NCH (CC / KC)       // 8 chunks

// NATTEN tiling: 14 queries/block, 7x22 key union padded to 160 keys
#define QB   14
#define NU   22
#define NK   (KS * NU)      // 154
#define NKP  160
#define NKT  (NKP / 16)     // 10 key tiles
#define KP   33             // K/Q row pad (coprime with 64 banks)
#define SP   165            // score row stride (odd)

typedef float v2f __attribute__((ext_vector_type(2)));
typedef float v8f __attribute__((ext_vector_type(8)));
typedef unsigned int u32x4 __attribute__((ext_vector_type(4)));
typedef int i32x4 __attribute__((ext_vector_type(4)));
typedef int i32x8 __attribute__((ext_vector_type(8)));

// ---------------------------------------------------------------------------
// 2D-tile TDM transfer (global -> LDS), CDNA5 D# layout.
// ---------------------------------------------------------------------------
__device__ __forceinline__ void tdm_load_2d(void* lds_ptr, const void* gptr,
                                            unsigned tile_d0, unsigned tile_d1,
                                            unsigned stride_d0)
{
  const unsigned lds_off = (unsigned)(unsigned long long)lds_ptr;
  const unsigned long long ga = (unsigned long long)gptr;

  u32x4 g0;
  g0[0] = 1u;                                                 // count=1
  g0[1] = lds_off;                                            // LDS byte address
  g0[2] = (unsigned)(ga & 0xFFFFFFFFu);                       // global_addr lo
  g0[3] = (unsigned)((ga >> 32) & 0x1FFFFFFu) | 0x80000000u;  // hi | type=2

  i32x8 g1;
  g1[0] = 0x00020000;                                         // data_size = 4B
  g1[1] = (int)((tile_d0 & 0xFFFFu) << 16);                   // tensor_dim0 lo
  g1[2] = (int)((tile_d0 >> 16) | ((tile_d1 & 0xFFFFu) << 16));
  g1[3] = (int)((tile_d1 >> 16) | (tile_d0 << 16));           // | tile_dim0
  g1[4] = (int)(tile_d1 & 0xFFFFu);                           // tile_dim1
  g1[5] = (int)stride_d0;                                     // row stride
  g1[6] = 0;
  g1[7] = 0;

  i32x4 g2 = {0, 0, 0, 0};
  i32x4 g3 = {0, 0, 0, 0};
#if defined(__clang_major__) && (__clang_major__ >= 23)
  i32x8 g4 = {0, 0, 0, 0, 0, 0, 0, 0};
  __builtin_amdgcn_tensor_load_to_lds(g0, g1, g2, g3, g4, 0);
#else
  __builtin_amdgcn_tensor_load_to_lds(g0, g1, g2, g3, 0);
#endif
}

// ---------------------------------------------------------------------------
// WMMA GEMM with TDM double-buffered LDS staging (unchanged from round 2).
// ---------------------------------------------------------------------------
template<int LDN, int MODE>
__global__ __launch_bounds__(256) void gemm_wmma_tdm(
    const float* __restrict__ A, const float* __restrict__ W,
    const float* __restrict__ bias,
    float* __restrict__ o0, float* __restrict__ o1, float* __restrict__ o2)
{
  __shared__ float sA[2][16 * KC];
  __shared__ float sB[2][KC * 128];

  const int lane = threadIdx.x & 31;
  const int wave = threadIdx.x >> 5;
  const int half = lane >> 4;
  const int l16  = lane & 15;
  const int nblock = blockIdx.x * 128;
  const int n0 = nblock + wave * 16;
  const int m0 = blockIdx.y * 16;

  const float* Abase = A + (size_t)m0 * CC;
  if (wave == 0) {
    tdm_load_2d(&sA[0][0], Abase,      KC,  16, CC);
    tdm_load_2d(&sB[0][0], W + nblock, 128, KC, LDN);
  }

  v8f acc = {};
  for (int c = 0; c < NCH; ++c) {
    if (wave == 0) {
      if (c + 1 < NCH) {
        const int kk = (c + 1) * KC;
        tdm_load_2d(&sA[(c + 1) & 1][0], Abase + kk,                    KC,  16, CC);
        tdm_load_2d(&sB[(c + 1) & 1][0], W + (size_t)kk * LDN + nblock, 128, KC, LDN);
        __builtin_amdgcn_s_wait_tensorcnt((short)2);
      } else {
        __builtin_amdgcn_s_wait_tensorcnt((short)0);
      }
    }
    __syncthreads();

    const float* sAc = &sA[c & 1][0];
    const float* sBc = &sB[c & 1][0];
    #pragma unroll
    for (int kk = 0; kk < KC; kk += 4) {
      const int ka = kk + half * 2;
      v2f av, bv;
      av.x = sAc[l16 * KC + ka];
      av.y = sAc[l16 * KC + ka + 1];
      bv.x = sBc[ka * 128 + wave * 16 + l16];
      bv.y = sBc[(ka + 1) * 128 + wave * 16 + l16];
      acc = __builtin_amdgcn_wmma_f32_16x16x4_f32(false, av, false, bv,
                                                  (short)0, acc, false, false);
    }
    __syncthreads();
  }

  const float bval = bias[n0 + l16];
  if (MODE == 0) {
    const int which = n0 >> 8;
    const int head  = (n0 & 255) >> 5;
    const int dbase = n0 & 31;
    float* dst = (which == 0) ? o0 : (which == 1) ? o1 : o2;
    const float scale = (which == 0) ? 0.17677669529663687f : 1.0f;
    #pragma unroll
    for (int r = 0; r < 8; ++r) {
      const int m    = m0 + r + half * 8;
      const int bidx = m / HW;
      const int hw   = m - bidx * HW;
      const float val = (acc[r] + bval) * scale;
      dst[((size_t)(bidx * NHD + head) * HW + hw) * DD + dbase + l16] = val;
    }
  } else {
    #pragma unroll
    for (int r = 0; r < 8; ++r) {
      const int m = m0 + r + half * 8;
      o0[(size_t)m * LDN + n0 + l16] = acc[r] + bval;
    }
  }
}

// ---------------------------------------------------------------------------
// WMMA 2D neighborhood attention.
// Block (4 waves) = (batch, head, row i, 14-query block).
// Scores S(16x160) = Q(16x32) . K^T via 10 key-tiles of wmma_f32_16x16x4;
// masked-bias softmax in LDS; out(16x32) = P(16x160) . V(160x32) via wmma,
// waves split (d-tile x K-half), partials combined through LDS.
// ---------------------------------------------------------------------------
__global__ __launch_bounds__(128) void natten_wmma_kernel(
    const float* __restrict__ qb, const float* __restrict__ kb,
    const float* __restrict__ vb, const float* __restrict__ rpb,
    float* __restrict__ ab)
{
  __shared__ float sK[NKP * KP];        // 21.1 KB, key-major rows padded to 33
  __shared__ float sV[NKP * DD];        // 20.5 KB
  __shared__ float sS[16 * SP];         // 10.6 KB scores -> probs
  __shared__ float sQ[16 * KP];         //  2.1 KB
  __shared__ float sRpb[RPB];
  __shared__ float sPart[2][16 * DD];   //  4.0 KB partial PV sums

  const int ix = blockIdx.x;
  const int i  = ix >> 2;               // image row
  const int jb = ix & 3;                // query block
  const int j0 = jb * QB;
  const int nh = blockIdx.y;
  const int b  = blockIdx.z;
  const int t  = threadIdx.x;
  const int lane = t & 31;
  const int wave = t >> 5;
  const int half = lane >> 4;
  const int l16  = lane & 15;

  int sh  = i - KS / 2;  if (sh  < 0) sh  = 0;  if (sh  > HH - KS) sh  = HH - KS;
  int swU = j0 - KS / 2; if (swU < 0) swU = 0;  if (swU > WW - KS) swU = WW - KS;

  const size_t headoff = (size_t)(b * NHD + nh) * HW * DD;
  const float* qbase = qb + headoff;
  const float* kbase = kb + headoff;
  const float* vbase = vb + headoff;

  // ---- stage Q (rows padded), rpb, and the 7x22 K/V union (padded to 160) ----
  for (int idx = t; idx < 16 * DD; idx += 128) {
    const int qi = idx >> 5, d = idx & 31;
    int qj = j0 + qi; if (qj > WW - 1) qj = WW - 1;        // clamp pad queries
    sQ[qi * KP + d] = qbase[((size_t)i * WW + qj) * DD + d];
  }
  const float* rp = rpb + nh * RPB;
  for (int idx = t; idx < RPB; idx += 128) sRpb[idx] = rp[idx];
  for (int idx = t; idx < NKP * DD; idx += 128) {
    const int kidx = idx >> 5, d = idx & 31;
    float kv = 0.0f, vv = 0.0f;
    if (kidx < NK) {
      const int a = kidx / NU;
      const int u = kidx - a * NU;
      int iw = swU + u; if (iw > WW - 1) iw = WW - 1;      // clamp, masked later
      const size_t goff = ((size_t)((sh + a) * WW + iw)) * DD + d;
      kv = kbase[goff];
      vv = vbase[goff];
    }
    sK[kidx * KP + d] = kv;
    sV[kidx * DD + d] = vv;
  }
  __syncthreads();

  // ---- scores: waves parallel over the 10 key tiles ----
  for (int kt = wave; kt < NKT; kt += 4) {
    v8f sacc = {};
    #pragma unroll
    for (int kk = 0; kk < DD; kk += 4) {
      const int ka = kk + half * 2;
      v2f av, bv;
      av.x = sQ[l16 * KP + ka];
      av.y = sQ[l16 * KP + ka + 1];
      bv.x = sK[(kt * 16 + l16) * KP + ka];
      bv.y = sK[(kt * 16 + l16) * KP + ka + 1];
      sacc = __builtin_amdgcn_wmma_f32_16x16x4_f32(false, av, false, bv,
                                                   (short)0, sacc, false, false);
    }
    #pragma unroll
    for (int r = 0; r < 8; ++r)
      sS[(r + half * 8) * SP + kt * 16 + l16] = sacc[r];
  }
  __syncthreads();

  // ---- mask + relative-position bias + softmax (thread-per-query) ----
  if (t < 16) {
    if (t < QB) {
      const int q = j0 + t;
      int swq = q - KS / 2; if (swq < 0) swq = 0; if (swq > WW - KS) swq = WW - KS;
      const int du = swq - swU;                   // valid u in [du, du+6]
      float mx = -1.0e30f;
      for (int a = 0; a < KS; ++a) {
        const int relh = sh + a - i + (KS - 1);
        const float* bh = &sRpb[relh * (2 * KS - 1)];
        for (int u = 0; u < NU; ++u) {
          const int idx = a * NU + u;
          float s = -1.0e30f;
          if (u >= du && u <= du + 6)
            s = sS[t * SP + idx] + bh[swU + u - q + (KS - 1)];
          sS[t * SP + idx] = s;
          mx = fmaxf(mx, s);
        }
      }
      float sum = 0.0f;
      for (int idx = 0; idx < NK; ++idx) {
        float p = __expf(sS[t * SP + idx] - mx);
        sS[t * SP + idx] = p;
        sum += p;
      }
      const float inv = 1.0f / sum;
      for (int idx = 0; idx < NK; ++idx) sS[t * SP + idx] *= inv;
      for (int idx = NK; idx < NKP; ++idx) sS[t * SP + idx] = 0.0f;
    } else {
      for (int idx = 0; idx < NKP; ++idx) sS[t * SP + idx] = 0.0f;  // pad rows
    }
  }
  __syncthreads();

  // ---- P @ V: wave -> (d-tile, K-half); 20 wmma each ----
  const int d0 = (wave & 1) * 16;
  const int kh = (wave >> 1) * (NKP / 2);
  v8f oacc = {};
  #pragma unroll
  for (int kk = 0; kk < NKP / 2; kk += 4) {
    const int ka = kh + kk + half * 2;
    v2f av, bv;
    av.x = sS[l16 * SP + ka];
    av.y = sS[l16 * SP + ka + 1];
    bv.x = sV[ka * DD + d0 + l16];
    bv.y = sV[(ka + 1) * DD + d0 + l16];
    oacc = __builtin_amdgcn_wmma_f32_16x16x4_f32(false, av, false, bv,
                                                 (short)0, oacc, false, false);
  }
  if (wave >= 2) {
    #pragma unroll
    for (int r = 0; r < 8; ++r)
      sPart[wave - 2][(r + half * 8) * DD + d0 + l16] = oacc[r];
  }
  __syncthreads();
  if (wave < 2) {
    #pragma unroll
    for (int r = 0; r < 8; ++r) {
      const int row = r + half * 8;
      if (row < QB) {                      // rows 14..15 are padding
        const float v = oacc[r] + sPart[wave][row * DD + d0 + l16];
        ab[((size_t)(b * HW + i * WW + j0 + row) * CC) + nh * DD + d0 + l16] = v;
      }
    }
  }
}

// ---------------------------------------------------------------------------
extern "C" void kernel_launch(void* const* d_in, const int* in_sizes, int n_in,
                              void* d_out, int out_size, void* d_ws, size_t ws_size,
                              hipStream_t stream)
{
  (void)in_sizes; (void)n_in; (void)out_size; (void)ws_size;
  const float* x    = (const float*)d_in[0];
  const float* Wqkv = (const float*)d_in[1];
  const float* bqkv = (const float*)d_in[2];
  const float* rpb  = (const float*)d_in[3];
  const float* Wp   = (const float*)d_in[4];
  const float* bp   = (const float*)d_in[5];
  float* out = (float*)d_out;

  float* ws = (float*)d_ws;
  const size_t SZ = (size_t)TOK * CC;
  float* qb = ws;
  float* kb = qb + SZ;
  float* vb = kb + SZ;
  float* ab = vb + SZ;

  gemm_wmma_tdm<N3C, 0><<<dim3(6, 392), 256, 0, stream>>>(x, Wqkv, bqkv, qb, kb, vb);
  natten_wmma_kernel   <<<dim3(HH * 4, NHD, BATCH), 128, 0, stream>>>(qb, kb, vb, rpb, ab);
  gemm_wmma_tdm<CC, 1><<<dim3(2, 392), 256, 0, stream>>>(ab, Wp, bp, out, nullptr, nullptr);
}